// SquaredGaussianMixture_47691316855485
// MI455X (gfx1250) — compile-verified
//
#include <hip/hip_runtime.h>
#include <hip/hip_bf16.h>
#include <math.h>

typedef float v2f __attribute__((ext_vector_type(2)));
typedef float v8f __attribute__((ext_vector_type(8)));

#define NB 1024   // blocks in main kernel (partials array size)
#define BT 256    // threads per block (8 wave32 waves)
#define KC 8      // clusters
// (2*pi)^4
#define TP4 1558.5454565440389f

// ---------- 4x4 inverse + det (Gauss-Jordan, no pivoting: inputs are SPD) ----
__device__ inline void inv4(const float* A, float* Ainv, float* detOut) {
  float M[16], I[16];
  for (int i = 0; i < 16; ++i) { M[i] = A[i]; I[i] = 0.f; }
  I[0] = I[5] = I[10] = I[15] = 1.f;
  float det = 1.f;
  for (int c = 0; c < 4; ++c) {
    float p = M[c*4+c];
    det *= p;
    float ip = 1.f / p;
    for (int j = 0; j < 4; ++j) { M[c*4+j] *= ip; I[c*4+j] *= ip; }
    for (int r = 0; r < 4; ++r) {
      if (r == c) continue;
      float f = M[r*4+c];
      for (int j = 0; j < 4; ++j) { M[r*4+j] -= f*M[c*4+j]; I[r*4+j] -= f*I[c*4+j]; }
    }
  }
  for (int i = 0; i < 16; ++i) Ainv[i] = I[i];
  *detOut = det;
}

// ---------- setup: coef[16][16], W[8][8], log z  (tiny, one thread) ----------
__global__ void sgm_setup(const float* __restrict__ means,
                          const float* __restrict__ chols,
                          const float* __restrict__ weights,
                          float* __restrict__ coef,
                          float* __restrict__ Wmat,
                          double* __restrict__ logz) {
  if (threadIdx.x != 0 || blockIdx.x != 0) return;
  float Sig[KC][16];
  for (int k = 0; k < KC; ++k) {
    // L = tril(chols[k]);  Sig = L L^T + I
    float L[16];
    for (int r = 0; r < 4; ++r)
      for (int c = 0; c < 4; ++c)
        L[r*4+c] = (c <= r) ? chols[k*16 + r*4 + c] : 0.f;
    for (int r = 0; r < 4; ++r)
      for (int c = 0; c < 4; ++c) {
        float s = (r == c) ? 1.f : 0.f;
        for (int t = 0; t < 4; ++t) s += L[r*4+t] * L[c*4+t];
        Sig[k][r*4+c] = s;
      }
    float Sinv[16], det;
    inv4(Sig[k], Sinv, &det);
    const float* mu = means + k*4;
    float b[4], cc = 0.f;
    for (int r = 0; r < 4; ++r) {
      float s = 0.f;
      for (int t = 0; t < 4; ++t) s += Sinv[r*4+t] * mu[t];
      b[r] = s;
      cc += s * mu[r];
    }
    float logn = -0.5f * logf(TP4 * det);
    // log dens = sum_f coef[f]*phi[f]
    // phi = [x0^2..x3^2, x0x1,x0x2,x0x3,x1x2,x1x3,x2x3, x0..x3, 1, 0]
    float* cr = coef + k*16;
    cr[0] = -0.5f*Sinv[0];  cr[1] = -0.5f*Sinv[5];  cr[2] = -0.5f*Sinv[10]; cr[3] = -0.5f*Sinv[15];
    cr[4] = -Sinv[1];  cr[5] = -Sinv[2];  cr[6] = -Sinv[3];
    cr[7] = -Sinv[6];  cr[8] = -Sinv[7];  cr[9] = -Sinv[11];
    cr[10] = b[0]; cr[11] = b[1]; cr[12] = b[2]; cr[13] = b[3];
    cr[14] = -0.5f*cc + logn;
    cr[15] = 0.f;
  }
  for (int k = KC; k < 16; ++k)           // padded cluster rows -> g = 0
    for (int j = 0; j < 16; ++j) coef[k*16+j] = 0.f;

  // W = softmax over all 64 w_i*w_j
  float p[64]; float mx = -INFINITY;
  for (int i = 0; i < KC; ++i)
    for (int j = 0; j < KC; ++j) {
      float v = weights[i]*weights[j];
      p[i*8+j] = v; if (v > mx) mx = v;
    }
  float ssum = 0.f;
  for (int i = 0; i < 64; ++i) { p[i] = expf(p[i]-mx); ssum += p[i]; }
  for (int i = 0; i < 64; ++i) Wmat[i] = p[i] / ssum;

  // z = sum_ij W_ij * N(mu_i; mu_j, Sig_i + Sig_j)
  double z = 0.0;
  for (int i = 0; i < KC; ++i)
    for (int j = 0; j < KC; ++j) {
      float Ss[16];
      for (int t = 0; t < 16; ++t) Ss[t] = Sig[i][t] + Sig[j][t];
      float Sinv[16], det;
      inv4(Ss, Sinv, &det);
      float dmu[4];
      for (int t = 0; t < 4; ++t) dmu[t] = means[i*4+t] - means[j*4+t];
      float md = 0.f;
      for (int r = 0; r < 4; ++r)
        for (int c = 0; c < 4; ++c) md += dmu[r]*Sinv[r*4+c]*dmu[c];
      float term = expf(-0.5f*md) * rsqrtf(TP4 * det);
      z += (double)(Wmat[i*8+j] * term);
    }
  *logz = log(z);
}

// ---------- main: stream X, WMMA log-densities, per-point quadratic form ----
// G(16x16) = Coef(16 clusters x 16 feat) x Phi^T(16 feat x 16 points) via 4
// chained v_wmma_f32_16x16x4_f32.  C/D layout => lane n (0..15) holds all 8
// cluster log-densities of point n in VGPRs 0..7; lanes 16..31 hold the
// zero-padded cluster rows (g == 0) and are masked out of the reduction.
// Loop control is forced scalar (readfirstlane) so EXEC stays all-ones at
// every WMMA, as the ISA requires, and the loop uses s_cbranch not exec masks.
__global__ void __launch_bounds__(BT)
sgm_main(const float4* __restrict__ X4,
         const float* __restrict__ coef,
         const float* __restrict__ Wmat,
         double* __restrict__ partials,
         int npoints, int ntiles) {
  const int lane = threadIdx.x & 31;
  const int m    = lane & 15;   // A: cluster row | B/D: point column
  const int h    = lane >> 4;   // lane half selects K sub-pair
  // wave index, forced into an SGPR so the grid-stride loop is uniform
  const int gwave  = __builtin_amdgcn_readfirstlane(
                       blockIdx.x * (blockDim.x >> 5) + (threadIdx.x >> 5));
  const int nwaves = gridDim.x * (blockDim.x >> 5);

  // loop-invariant A fragments: afrag[kk] = {coef[m][4kk+2h], coef[m][4kk+2h+1]}
  v2f afrag[4];
#pragma unroll
  for (int kk = 0; kk < 4; ++kk) {
    int base = m*16 + kk*4 + h*2;
    afrag[kk].x = coef[base];
    afrag[kk].y = coef[base+1];
  }
  // pairwise softmax weights (wave-uniform -> SGPRs)
  float Wl[64];
#pragma unroll
  for (int i = 0; i < 64; ++i) Wl[i] = Wmat[i];

  double acc = 0.0;
  for (int t = gwave; t < ntiles; t += nwaves) {
    int pidx = t*16 + m;
    int pclamp = pidx < npoints ? pidx : (npoints - 1);
    float4 x = X4[pclamp];                       // global_load_b128
    // branchless prefetch of this wave's next tile (clamped, speculative)
    int tn = t + nwaves; if (tn >= ntiles) tn = ntiles - 1;
    __builtin_prefetch(&X4[tn*16 + m], 0, 0);    // global_prefetch_b8

    // 16 features of this lane's point
    float f0 = x.x*x.x, f1 = x.y*x.y, f2 = x.z*x.z, f3 = x.w*x.w;
    float f4 = x.x*x.y, f5 = x.x*x.z, f6 = x.x*x.w;
    float f7 = x.y*x.z, f8 = x.y*x.w, f9 = x.z*x.w;
    float f10 = x.x, f11 = x.y, f12 = x.z, f13 = x.w, f14 = 1.f, f15 = 0.f;

    // B fragments: b_kk = {phi[4kk+2h], phi[4kk+2h+1]}
    v2f b0, b1, b2, b3;
    b0.x = h ? f2  : f0;   b0.y = h ? f3  : f1;
    b1.x = h ? f6  : f4;   b1.y = h ? f7  : f5;
    b2.x = h ? f10 : f8;   b2.y = h ? f11 : f9;
    b3.x = h ? f14 : f12;  b3.y = h ? f15 : f13;

    v8f c = {0.f,0.f,0.f,0.f,0.f,0.f,0.f,0.f};
    c = __builtin_amdgcn_wmma_f32_16x16x4_f32(false, afrag[0], false, b0, (short)0, c, false, false);
    c = __builtin_amdgcn_wmma_f32_16x16x4_f32(false, afrag[1], false, b1, (short)0, c, false, false);
    c = __builtin_amdgcn_wmma_f32_16x16x4_f32(false, afrag[2], false, b2, (short)0, c, false, false);
    c = __builtin_amdgcn_wmma_f32_16x16x4_f32(false, afrag[3], false, b3, (short)0, c, false, false);

    // dens_k = exp(g_k); lanes 0..15: VGPR r == cluster r of point (t*16+lane)
    float dv[8];
#pragma unroll
    for (int r = 0; r < 8; ++r) dv[r] = __expf(c[r]);

    // num = dens^T W dens  (W lives in SGPRs)
    float num = 0.f;
#pragma unroll
    for (int i = 0; i < 8; ++i) {
      float ti = 0.f;
#pragma unroll
      for (int j = 0; j < 8; ++j) ti = fmaf(Wl[i*8+j], dv[j], ti);
      num = fmaf(ti, dv[i], num);
    }
    bool valid = (h == 0) && (pidx < npoints);
    acc += valid ? (double)num : 0.0;
  }

  // deterministic block tree reduction
  __shared__ double sdata[BT];
  sdata[threadIdx.x] = acc;
  __syncthreads();
  for (int s = BT/2; s > 0; s >>= 1) {
    if (threadIdx.x < s) sdata[threadIdx.x] += sdata[threadIdx.x + s];
    __syncthreads();
  }
  if (threadIdx.x == 0) partials[blockIdx.x] = sdata[0];
}

// ---------- finalize: sequential (deterministic) sum + log ------------------
__global__ void sgm_finalize(const double* __restrict__ partials, int nblocks,
                             const double* __restrict__ logz,
                             float* __restrict__ out, int npoints) {
  if (threadIdx.x == 0 && blockIdx.x == 0) {
    double s = 0.0;
    for (int i = 0; i < nblocks; ++i) s += partials[i];
    out[0] = (float)(-(log(s) - *logz) / (double)npoints);
  }
}

extern "C" void kernel_launch(void* const* d_in, const int* in_sizes, int n_in,
                              void* d_out, int out_size, void* d_ws, size_t ws_size,
                              hipStream_t stream) {
  const float* X       = (const float*)d_in[0];  // [N,4]
  const float* means   = (const float*)d_in[1];  // [8,4]
  const float* chols   = (const float*)d_in[2];  // [8,4,4]
  const float* weights = (const float*)d_in[3];  // [8]
  // d_in[4] = it (unused by the reference computation)

  int npoints = in_sizes[0] / 4;
  int ntiles  = (npoints + 15) / 16;

  char* ws = (char*)d_ws;
  double* partials = (double*)ws;                              // NB doubles
  double* logz     = (double*)(ws + NB*sizeof(double));        // 1 double
  float*  coef     = (float*)(ws + NB*sizeof(double) + 16);    // 256 floats
  float*  Wmat     = coef + 256;                               // 64 floats

  sgm_setup<<<1, 64, 0, stream>>>(means, chols, weights, coef, Wmat, logz);
  sgm_main<<<NB, BT, 0, stream>>>((const float4*)X, coef, Wmat, partials,
                                  npoints, ntiles);
  sgm_finalize<<<1, 64, 0, stream>>>(partials, NB, logz, (float*)d_out, npoints);
}